// FPModule_26809185861898
// MI455X (gfx1250) — compile-verified
//
#include <hip/hip_runtime.h>
#include <math.h>

typedef __attribute__((ext_vector_type(2))) float v2f;
typedef __attribute__((ext_vector_type(8))) float v8f;

#define B_    16
#define N_    4096
#define M_    1024
#define C1_   256
#define C2_   512
#define VD_   64
#define CIN0  832      // 512 + 256 + 64
#define CMID  512
#define COUT_ 256
#define P_    (B_ * N_)   // 65536 points

// ---------------------------------------------------------------------------
// 1. three_nn: per query point, 3 nearest of 1024 known points (squared dist)
// ---------------------------------------------------------------------------
__global__ __launch_bounds__(256) void knn_kernel(
    const float* __restrict__ unknown, const float* __restrict__ known,
    int* __restrict__ idxOut, float* __restrict__ wOut)
{
  __shared__ float kx[M_], ky[M_], kz[M_];
  const int tid = threadIdx.x;
  const int blocksPerB = N_ / 256;            // 16
  const int b  = blockIdx.x / blocksPerB;
  const int n0 = (blockIdx.x % blocksPerB) * 256;
  for (int i = tid; i < M_; i += 256) {
    const float* kp = known + ((size_t)b * M_ + i) * 3;
    kx[i] = kp[0]; ky[i] = kp[1]; kz[i] = kp[2];
  }
  __syncthreads();
  const int n = n0 + tid;
  const float* up = unknown + ((size_t)b * N_ + n) * 3;
  const float ux = up[0], uy = up[1], uz = up[2];
  float d0 = 3.4e38f, d1 = 3.4e38f, d2 = 3.4e38f;
  int   i0 = 0, i1 = 0, i2 = 0;
  for (int m = 0; m < M_; ++m) {
    const float dx = ux - kx[m], dy = uy - ky[m], dz = uz - kz[m];
    const float d  = dx*dx + dy*dy + dz*dz;
    if (d < d0)      { d2=d1; i2=i1; d1=d0; i1=i0; d0=d; i0=m; }
    else if (d < d1) { d2=d1; i2=i1; d1=d;  i1=m; }
    else if (d < d2) { d2=d;  i2=m; }
  }
  const float r0 = 1.f/(d0+1e-8f), r1 = 1.f/(d1+1e-8f), r2 = 1.f/(d2+1e-8f);
  const float rs = 1.f/(r0+r1+r2);
  const size_t p = ((size_t)b * N_ + n) * 3;
  idxOut[p+0]=i0; idxOut[p+1]=i1; idxOut[p+2]=i2;
  wOut[p+0]=r0*rs; wOut[p+1]=r1*rs; wOut[p+2]=r2*rs;
}

// ---------------------------------------------------------------------------
// 2. transpose known_feats [B,512,1024] -> known_t [B,1024,512]
// ---------------------------------------------------------------------------
__global__ __launch_bounds__(256) void transpose_known(
    const float* __restrict__ in, float* __restrict__ out)
{
  __shared__ float t[32][33];
  const int m0 = blockIdx.x * 32;
  const int c0 = blockIdx.y * 32;
  const int b  = blockIdx.z;
  const int l  = threadIdx.x & 31;
  const int r  = threadIdx.x >> 5;            // 0..7
  const float* src = in + (size_t)b * C2_ * M_;
  for (int rr = r; rr < 32; rr += 8)
    t[rr][l] = src[(size_t)(c0 + rr) * M_ + m0 + l];
  __syncthreads();
  float* dst = out + (size_t)b * M_ * C2_;
  for (int rr = r; rr < 32; rr += 8)
    dst[(size_t)(m0 + rr) * C2_ + c0 + l] = t[l][rr];
}

// ---------------------------------------------------------------------------
// 3. global max-pool over n: g[b,c] = max_n uf[b,c,n]
// ---------------------------------------------------------------------------
__global__ __launch_bounds__(256) void maxpool_kernel(
    const float* __restrict__ uf, float* __restrict__ g)
{
  __shared__ float red[256];
  const int b = blockIdx.x >> 8;              // / C1_
  const int c = blockIdx.x & (C1_ - 1);
  const float* row = uf + ((size_t)b * C1_ + c) * N_;
  float m = -3.4e38f;
  for (int i = threadIdx.x; i < N_; i += 256) m = fmaxf(m, row[i]);
  red[threadIdx.x] = m;
  __syncthreads();
  for (int s = 128; s > 0; s >>= 1) {
    if (threadIdx.x < s) red[threadIdx.x] = fmaxf(red[threadIdx.x], red[threadIdx.x + s]);
    __syncthreads();
  }
  if (threadIdx.x == 0) g[b * C1_ + c] = red[0];
}

// ---------------------------------------------------------------------------
// 4. squeeze-excite: s = sigmoid(w2 @ relu(w1 @ g))
// ---------------------------------------------------------------------------
__global__ __launch_bounds__(256) void se_kernel(
    const float* __restrict__ g, const float* __restrict__ w1,
    const float* __restrict__ w2, float* __restrict__ s)
{
  __shared__ float gl[C1_];
  __shared__ float hl[64];
  const int b = blockIdx.x, tid = threadIdx.x;
  gl[tid] = g[b * C1_ + tid];
  __syncthreads();
  if (tid < 64) {
    float a = 0.f;
    for (int c = 0; c < C1_; ++c) a += w1[tid * C1_ + c] * gl[c];
    hl[tid] = fmaxf(a, 0.f);
  }
  __syncthreads();
  float a = 0.f;
  for (int o = 0; o < 64; ++o) a += w2[tid * 64 + o] * hl[o];
  s[b * C1_ + tid] = 1.f / (1.f + expf(-a));
}

// ---------------------------------------------------------------------------
// 5. assemble X[p][832] = [interp(512) | s*uf(256) | vd tiled(64)], point-major
// ---------------------------------------------------------------------------
__global__ __launch_bounds__(256) void assemble_kernel(
    const float* __restrict__ knownT, const int* __restrict__ idx,
    const float* __restrict__ wgt, const float* __restrict__ uf,
    const float* __restrict__ vd, const float* __restrict__ s,
    float* __restrict__ X)
{
  __shared__ float sL[C1_];
  __shared__ int   iL[64 * 3];
  __shared__ float wL[64 * 3];
  __shared__ float tile[64][65];
  const int tid = threadIdx.x;
  const int b   = blockIdx.x >> 6;            // N_/64 = 64 tiles per batch
  const int t64 = blockIdx.x & 63;
  const int n0  = t64 * 64;
  const size_t p0 = (size_t)b * N_ + n0;

  sL[tid] = s[b * C1_ + tid];
  if (tid < 192) {
    iL[tid] = idx[p0 * 3 + tid];
    wL[tid] = wgt[p0 * 3 + tid];
  }
  __syncthreads();

  // interpolation -> X[p][0..511]; prefetch next point's rows while computing
  const float* kb = knownT + (size_t)b * M_ * C2_;
  for (int j = 0; j < 64; ++j) {
    const float a0 = wL[j*3+0], a1 = wL[j*3+1], a2 = wL[j*3+2];
    const float* r0 = kb + (size_t)iL[j*3+0] * C2_;
    const float* r1 = kb + (size_t)iL[j*3+1] * C2_;
    const float* r2 = kb + (size_t)iL[j*3+2] * C2_;
    if (j + 1 < 64) {
      const float* n0p = kb + (size_t)iL[(j+1)*3+0] * C2_;
      const float* n1p = kb + (size_t)iL[(j+1)*3+1] * C2_;
      const float* n2p = kb + (size_t)iL[(j+1)*3+2] * C2_;
      __builtin_prefetch(n0p + tid, 0, 0);    // global_prefetch_b8
      __builtin_prefetch(n1p + tid, 0, 0);
      __builtin_prefetch(n2p + tid, 0, 0);
    }
    float* xr = X + (p0 + j) * CIN0;
    for (int c = tid; c < C2_; c += 256)
      xr[c] = a0 * r0[c] + a1 * r1[c] + a2 * r2[c];
  }

  const int jl = tid & 63, q = tid >> 6;      // q = 0..3
  // s * uf -> X[p][512..767] via 64x64 LDS transpose
  for (int cblk = 0; cblk < 4; ++cblk) {
    __syncthreads();
    for (int rr = q; rr < 64; rr += 4)
      tile[rr][jl] = uf[((size_t)b * C1_ + cblk * 64 + rr) * N_ + n0 + jl];
    __syncthreads();
    const int cl = tid & 63;
    for (int rr = q; rr < 64; rr += 4) {
      const int c = cblk * 64 + cl;
      X[(p0 + rr) * CIN0 + C2_ + c] = sL[c] * tile[cl][rr];
    }
  }

  // vd tiled (n % 1024) -> X[p][768..831]
  const int m0 = n0 & (M_ - 1);
  __syncthreads();
  for (int rr = q; rr < 64; rr += 4)
    tile[rr][jl] = vd[((size_t)b * VD_ + rr) * M_ + m0 + jl];
  __syncthreads();
  {
    const int vl = tid & 63;
    for (int rr = q; rr < 64; rr += 4)
      X[(p0 + rr) * CIN0 + C2_ + C1_ + vl] = tile[vl][rr];
  }
}

// ---------------------------------------------------------------------------
// 6/7. GEMM + BN + ReLU via V_WMMA_F32_16X16X4_F32 (fp32 matrix pipe)
//   out[p,o] = relu( (sum_c in[p,c]*W[o,c]) * scale[o] + shift[o] )
//   wave = 32 points x 64 outputs (2 A tiles x 4 B tiles = 8 accumulators).
//   Register double-buffering: next step's 6 fragments are issued before the
//   8 WMMAs of the current step, so the matrix pipe covers the load latency.
// ---------------------------------------------------------------------------
__global__ __launch_bounds__(256) void gemm_bn_relu(
    const float* __restrict__ Xin, const float* __restrict__ W,
    const float* __restrict__ bn_g, const float* __restrict__ bn_b,
    const float* __restrict__ bn_m, const float* __restrict__ bn_v,
    float* __restrict__ out, int CIN, int COUT, int outBCN)
{
  const int tid  = threadIdx.x;
  const int wave = tid >> 5;
  const int lane = tid & 31;
  const int lh   = lane & 15;
  const int hi   = lane >> 4;
  const size_t p0 = (size_t)blockIdx.x * 256 + wave * 32;
  const int o0 = blockIdx.y * 64;

  const float* ar0 = Xin + (p0 +  0 + lh) * (size_t)CIN + 2 * hi;
  const float* ar1 = Xin + (p0 + 16 + lh) * (size_t)CIN + 2 * hi;
  const float* br0 = W + (size_t)(o0 +  0 + lh) * CIN + 2 * hi;
  const float* br1 = W + (size_t)(o0 + 16 + lh) * CIN + 2 * hi;
  const float* br2 = W + (size_t)(o0 + 32 + lh) * CIN + 2 * hi;
  const float* br3 = W + (size_t)(o0 + 48 + lh) * CIN + 2 * hi;

  v8f acc[2][4];
  #pragma unroll
  for (int i = 0; i < 2; ++i)
    #pragma unroll
    for (int t = 0; t < 4; ++t)
      acc[i][t] = (v8f){0.f,0.f,0.f,0.f,0.f,0.f,0.f,0.f};

  // prologue: fragments for k = 0
  v2f a0 = *(const v2f*)(ar0);
  v2f a1 = *(const v2f*)(ar1);
  v2f w0 = *(const v2f*)(br0);
  v2f w1 = *(const v2f*)(br1);
  v2f w2 = *(const v2f*)(br2);
  v2f w3 = *(const v2f*)(br3);

  for (int k = 0; k < CIN - 4; k += 4) {
    // issue next step's loads before consuming current fragments
    const v2f na0 = *(const v2f*)(ar0 + k + 4);
    const v2f na1 = *(const v2f*)(ar1 + k + 4);
    const v2f nw0 = *(const v2f*)(br0 + k + 4);
    const v2f nw1 = *(const v2f*)(br1 + k + 4);
    const v2f nw2 = *(const v2f*)(br2 + k + 4);
    const v2f nw3 = *(const v2f*)(br3 + k + 4);

    acc[0][0] = __builtin_amdgcn_wmma_f32_16x16x4_f32(false, a0, false, w0, (short)0, acc[0][0], false, false);
    acc[0][1] = __builtin_amdgcn_wmma_f32_16x16x4_f32(false, a0, false, w1, (short)0, acc[0][1], false, false);
    acc[0][2] = __builtin_amdgcn_wmma_f32_16x16x4_f32(false, a0, false, w2, (short)0, acc[0][2], false, false);
    acc[0][3] = __builtin_amdgcn_wmma_f32_16x16x4_f32(false, a0, false, w3, (short)0, acc[0][3], false, false);
    acc[1][0] = __builtin_amdgcn_wmma_f32_16x16x4_f32(false, a1, false, w0, (short)0, acc[1][0], false, false);
    acc[1][1] = __builtin_amdgcn_wmma_f32_16x16x4_f32(false, a1, false, w1, (short)0, acc[1][1], false, false);
    acc[1][2] = __builtin_amdgcn_wmma_f32_16x16x4_f32(false, a1, false, w2, (short)0, acc[1][2], false, false);
    acc[1][3] = __builtin_amdgcn_wmma_f32_16x16x4_f32(false, a1, false, w3, (short)0, acc[1][3], false, false);

    a0 = na0; a1 = na1; w0 = nw0; w1 = nw1; w2 = nw2; w3 = nw3;
  }
  // final k step
  acc[0][0] = __builtin_amdgcn_wmma_f32_16x16x4_f32(false, a0, false, w0, (short)0, acc[0][0], false, false);
  acc[0][1] = __builtin_amdgcn_wmma_f32_16x16x4_f32(false, a0, false, w1, (short)0, acc[0][1], false, false);
  acc[0][2] = __builtin_amdgcn_wmma_f32_16x16x4_f32(false, a0, false, w2, (short)0, acc[0][2], false, false);
  acc[0][3] = __builtin_amdgcn_wmma_f32_16x16x4_f32(false, a0, false, w3, (short)0, acc[0][3], false, false);
  acc[1][0] = __builtin_amdgcn_wmma_f32_16x16x4_f32(false, a1, false, w0, (short)0, acc[1][0], false, false);
  acc[1][1] = __builtin_amdgcn_wmma_f32_16x16x4_f32(false, a1, false, w1, (short)0, acc[1][1], false, false);
  acc[1][2] = __builtin_amdgcn_wmma_f32_16x16x4_f32(false, a1, false, w2, (short)0, acc[1][2], false, false);
  acc[1][3] = __builtin_amdgcn_wmma_f32_16x16x4_f32(false, a1, false, w3, (short)0, acc[1][3], false, false);

  #pragma unroll
  for (int t = 0; t < 4; ++t) {
    const int ch = o0 + t * 16 + lh;
    const float sc = bn_g[ch] * rsqrtf(bn_v[ch] + 1e-5f);
    const float sh = bn_b[ch] - bn_m[ch] * sc;
    #pragma unroll
    for (int pt = 0; pt < 2; ++pt) {
      const v8f a = acc[pt][t];
      #pragma unroll
      for (int r = 0; r < 8; ++r) {
        const float y = fmaxf(a[r] * sc + sh, 0.f);
        const size_t p = p0 + pt * 16 + r + 8 * hi; // D: lane<16 M=r, lane>=16 M=r+8
        if (outBCN) {
          const size_t bb = p >> 12;                // / N_
          const size_t n  = p & (N_ - 1);
          out[(bb * COUT + ch) * N_ + n] = y;       // [B, COUT, N]
        } else {
          out[p * (size_t)COUT + ch] = y;           // [P, COUT] point-major
        }
      }
    }
  }
}

// ---------------------------------------------------------------------------
extern "C" void kernel_launch(void* const* d_in, const int* in_sizes, int n_in,
                              void* d_out, int out_size, void* d_ws, size_t ws_size,
                              hipStream_t stream) {
  (void)in_sizes; (void)n_in; (void)out_size; (void)ws_size;
  const float* unknown = (const float*)d_in[0];
  const float* known   = (const float*)d_in[1];
  const float* uf      = (const float*)d_in[2];
  const float* kf      = (const float*)d_in[3];
  const float* vd      = (const float*)d_in[4];
  const float* lab_w1  = (const float*)d_in[5];
  const float* lab_w2  = (const float*)d_in[6];
  const float* cw0     = (const float*)d_in[7];
  const float* g0      = (const float*)d_in[8];
  const float* b0      = (const float*)d_in[9];
  const float* m0      = (const float*)d_in[10];
  const float* v0      = (const float*)d_in[11];
  const float* cw1     = (const float*)d_in[12];
  const float* g1      = (const float*)d_in[13];
  const float* b1      = (const float*)d_in[14];
  const float* m1      = (const float*)d_in[15];
  const float* v1      = (const float*)d_in[16];
  float* out = (float*)d_out;

  float* wsf    = (float*)d_ws;
  int*   idxW   = (int*)wsf;                               // P*3 ints
  float* wgtW   = wsf + (size_t)P_ * 3;                    // P*3
  float* knownT = wgtW + (size_t)P_ * 3;                   // B*M*C2
  float* gW     = knownT + (size_t)B_ * M_ * C2_;          // B*C1
  float* sW     = gW + B_ * C1_;                           // B*C1
  float* X      = sW + B_ * C1_;                           // P*832
  float* Y0     = X + (size_t)P_ * CIN0;                   // P*512

  knn_kernel<<<B_ * (N_ / 256), 256, 0, stream>>>(unknown, known, idxW, wgtW);
  transpose_known<<<dim3(M_ / 32, C2_ / 32, B_), 256, 0, stream>>>(kf, knownT);
  maxpool_kernel<<<B_ * C1_, 256, 0, stream>>>(uf, gW);
  se_kernel<<<B_, 256, 0, stream>>>(gW, lab_w1, lab_w2, sW);
  assemble_kernel<<<B_ * (N_ / 64), 256, 0, stream>>>(knownT, idxW, wgtW, uf, vd, sW, X);
  gemm_bn_relu<<<dim3(P_ / 256, CMID / 64), 256, 0, stream>>>(
      X, cw0, g0, b0, m0, v0, Y0, CIN0, CMID, 0);
  gemm_bn_relu<<<dim3(P_ / 256, COUT_ / 64), 256, 0, stream>>>(
      Y0, cw1, g1, b1, m1, v1, out, CMID, COUT_, 1);
}